// GAT_46772193853510
// MI455X (gfx1250) — compile-verified
//
#include <hip/hip_runtime.h>
#include <math.h>

#define N_NODES 50000
#define F_IN    512
#define F1      128      // H1*C1
#define H1_     8
#define C1_     16
#define C2_     40
#define NEG_SLOPE 0.2f

#define KC      128      // K-chunk staged in LDS for GEMM1
#define LDP     (KC + 4) // padded LDS row stride (132 floats) -> conflict-free ds_load_b64

typedef float v2f __attribute__((ext_vector_type(2)));
typedef float v8f __attribute__((ext_vector_type(8)));

// ---- monotone float<->uint encoding for exact atomic float max ----
__device__ __forceinline__ unsigned f2ord(float f) {
    int i = __float_as_int(f);
    return (i >= 0) ? ((unsigned)i ^ 0x80000000u) : ~(unsigned)i;
}
__device__ __forceinline__ float ord2f(unsigned k) {
    int i = (k & 0x80000000u) ? (int)(k ^ 0x80000000u) : (int)~k;
    return __int_as_float(i);
}
#define ORD_NEG_INF 0x007FFFFFu   // f2ord(-inf)

__global__ void init_u32(unsigned* __restrict__ p, unsigned v, int n) {
    int i = blockIdx.x * blockDim.x + threadIdx.x;
    if (i < n) p[i] = v;
}

// ============ GEMM1: h1[N,128] = x[N,512] @ W1[512,128]  (fp32 WMMA) ============
// Block = 256 threads (8 waves) computes a 64x128 macro-tile.
// A staged in LDS per 64x128 K-chunk; each wave holds 4 accumulators (4 M-tiles)
// so one B load feeds 4 WMMAs. Grid = ceil(3125/4) = 782 blocks (rows clamped).
__global__ __launch_bounds__(256) void gemm1_wmma(const float* __restrict__ X,
                                                  const float* __restrict__ W,
                                                  float* __restrict__ Hout) {
    __shared__ float As[64 * LDP];                 // 33,792 bytes
    const int tid   = threadIdx.x;
    const int lane  = tid & 31;
    const int wave  = tid >> 5;                    // 0..7 = N tile
    const int R0    = blockIdx.x * 64;             // first row of macro-tile
    const int n     = (wave << 4) + (lane & 15);   // output column
    const int khalf = (lane >> 4) << 1;            // 0 or 2 per ISA 16x4 A layout
    const int mr    = lane & 15;

    v8f c0 = {}, c1 = {}, c2 = {}, c3 = {};

    for (int kc = 0; kc < F_IN; kc += KC) {
        __syncthreads();
        // ---- cooperative stage: 64 rows x 128 cols, float4 per transfer ----
        #pragma unroll
        for (int i = 0; i < 8; ++i) {              // 2048 float4 / 256 threads
            int idx4 = tid + (i << 8);
            int r    = idx4 >> 5;                  // 32 float4 per row
            int cc   = (idx4 & 31) << 2;
            int gr   = R0 + r;  gr = gr < N_NODES ? gr : N_NODES - 1;
            float4 v = *(const float4*)(X + (size_t)gr * F_IN + kc + cc);
            *(float4*)(&As[r * LDP + cc]) = v;
        }
        __syncthreads();

        #pragma unroll 4
        for (int k0 = 0; k0 < KC; k0 += 4) {
            v2f b;  // B 4x16: rows khalf/khalf+1 striped across lanes
            b.x = W[(size_t)(kc + k0 + khalf) * F1 + n];
            b.y = W[(size_t)(kc + k0 + khalf + 1) * F1 + n];
            v2f a0, a1, a2, a3;                    // ds_load_b64, bank-conflict free
            a0.x = As[(mr)      * LDP + k0 + khalf]; a0.y = As[(mr)      * LDP + k0 + khalf + 1];
            a1.x = As[(mr + 16) * LDP + k0 + khalf]; a1.y = As[(mr + 16) * LDP + k0 + khalf + 1];
            a2.x = As[(mr + 32) * LDP + k0 + khalf]; a2.y = As[(mr + 32) * LDP + k0 + khalf + 1];
            a3.x = As[(mr + 48) * LDP + k0 + khalf]; a3.y = As[(mr + 48) * LDP + k0 + khalf + 1];
            c0 = __builtin_amdgcn_wmma_f32_16x16x4_f32(false, a0, false, b, (short)0, c0, false, false);
            c1 = __builtin_amdgcn_wmma_f32_16x16x4_f32(false, a1, false, b, (short)0, c1, false, false);
            c2 = __builtin_amdgcn_wmma_f32_16x16x4_f32(false, a2, false, b, (short)0, c2, false, false);
            c3 = __builtin_amdgcn_wmma_f32_16x16x4_f32(false, a3, false, b, (short)0, c3, false, false);
        }
    }
    // ---- store: VGPR r -> row = base + r + 8*(lane>=16), col n ----
    const int rb = ((lane >> 4) << 3);
    #pragma unroll
    for (int t = 0; t < 4; ++t) {
        const v8f* cp = (t == 0) ? &c0 : (t == 1) ? &c1 : (t == 2) ? &c2 : &c3;
        #pragma unroll
        for (int r = 0; r < 8; ++r) {
            int row = R0 + (t << 4) + rb + r;
            if (row < N_NODES)
                Hout[(size_t)row * F1 + n] = (*cp)[r];
        }
    }
}

// ============ GEMM2: h2[N,40] = act[N,128] @ W2[128,40]  (fp32 WMMA, N padded to 48) ============
__global__ __launch_bounds__(96) void gemm2_wmma(const float* __restrict__ A,
                                                 const float* __restrict__ W,
                                                 float* __restrict__ Hout) {
    const int lane  = threadIdx.x & 31;
    const int wave  = threadIdx.x >> 5;            // 0..2 = N tile
    const int mt    = blockIdx.x;
    const int m     = (mt << 4) + (lane & 15);
    const int n     = (wave << 4) + (lane & 15);   // 0..47
    const int khalf = (lane >> 4) << 1;
    const float wmask = (n < C2_) ? 1.0f : 0.0f;   // mask padded cols (no EXEC divergence)
    const int   nc    = (n < C2_) ? n : 0;         // clamp to avoid OOB reads
    const float* arow = A + (size_t)m * F1;
    v8f c = {};
    #pragma unroll 4
    for (int k0 = 0; k0 < F1; k0 += 4) {
        v2f a;
        a.x = arow[k0 + khalf];
        a.y = arow[k0 + khalf + 1];
        v2f b;
        b.x = W[(size_t)(k0 + khalf) * C2_ + nc] * wmask;
        b.y = W[(size_t)(k0 + khalf + 1) * C2_ + nc] * wmask;
        c = __builtin_amdgcn_wmma_f32_16x16x4_f32(false, a, false, b, (short)0, c, false, false);
    }
    if (n < C2_) {
        const int mbase = (mt << 4) + ((lane >> 4) << 3);
        #pragma unroll
        for (int r = 0; r < 8; ++r)
            Hout[(size_t)(mbase + r) * C2_ + n] = c[r];
    }
}

// ============ per-node attention scores: a_src[n,h] = <h[n,h,:], att_src[h,:]> ============
__global__ void att_scores(const float* __restrict__ h,
                           const float* __restrict__ att_s,
                           const float* __restrict__ att_d,
                           float* __restrict__ as_, float* __restrict__ ad_,
                           int H, int C) {
    int idx = blockIdx.x * blockDim.x + threadIdx.x;  // n*H + hh
    if (idx >= N_NODES * H) return;
    int hh = idx % H;
    const float* row = h + (size_t)(idx / H) * (H * C) + hh * C;
    float s = 0.f, d = 0.f;
    for (int c = 0; c < C; ++c) {
        float v = row[c];
        s += v * att_s[hh * C + c];
        d += v * att_d[hh * C + c];
    }
    as_[idx] = s;
    ad_[idx] = d;
}

__device__ __forceinline__ void edge_ids(const long long* __restrict__ ei, int E,
                                         int e, int& s, int& d) {
    if (e < E) { s = (int)ei[e]; d = (int)ei[(size_t)E + e]; }
    else       { s = d = e - E; }   // self loop
}

// ============ pass A: segment max of leaky_relu(a_src[src]+a_dst[dst]) over dst ============
__global__ void edge_max(const long long* __restrict__ ei, int E, int ET,
                         const float* __restrict__ as_, const float* __restrict__ ad_,
                         unsigned* __restrict__ m, int H) {
    int total = ET * H;
    for (int idx = blockIdx.x * blockDim.x + threadIdx.x; idx < total;
         idx += gridDim.x * blockDim.x) {
        int e = idx / H, hh = idx % H;
        int s, d; edge_ids(ei, E, e, s, d);
        float v = as_[s * H + hh] + ad_[d * H + hh];
        v = (v < 0.f) ? v * NEG_SLOPE : v;
        atomicMax(&m[d * H + hh], f2ord(v));
    }
}

// ============ pass B: denom[dst,h] += exp(e - m[dst,h]) ============
__global__ void edge_expsum(const long long* __restrict__ ei, int E, int ET,
                            const float* __restrict__ as_, const float* __restrict__ ad_,
                            const unsigned* __restrict__ m, float* __restrict__ denom,
                            int H) {
    int total = ET * H;
    for (int idx = blockIdx.x * blockDim.x + threadIdx.x; idx < total;
         idx += gridDim.x * blockDim.x) {
        int e = idx / H, hh = idx % H;
        int s, d; edge_ids(ei, E, e, s, d);
        float v = as_[s * H + hh] + ad_[d * H + hh];
        v = (v < 0.f) ? v * NEG_SLOPE : v;
        float ex = __expf(v - ord2f(m[d * H + hh]));
        atomicAdd(&denom[d * H + hh], ex);
    }
}

// ============ pass C: out[dst,:] += alpha * h[src,:]  (float4 gather, recompute alpha) ============
// One thread per (edge, 4-channel group). C % 4 == 0 for both layers, so a group
// never straddles heads.
__global__ void edge_aggregate4(const long long* __restrict__ ei, int E, int ET,
                                const float* __restrict__ as_, const float* __restrict__ ad_,
                                const unsigned* __restrict__ m, const float* __restrict__ denom,
                                const float* __restrict__ h, float* __restrict__ out,
                                int H, int C) {
    const int F  = H * C;
    const int F4 = F >> 2;
    int total = ET * F4;   // layer1: 850k*32 = 27.2M
    for (int idx = blockIdx.x * blockDim.x + threadIdx.x; idx < total;
         idx += gridDim.x * blockDim.x) {
        int e  = idx / F4;
        int j  = (idx - e * F4) << 2;
        int hh = j / C;
        int s, d; edge_ids(ei, E, e, s, d);
        float v = as_[s * H + hh] + ad_[d * H + hh];
        v = (v < 0.f) ? v * NEG_SLOPE : v;
        float alpha = __expf(v - ord2f(m[d * H + hh])) / (denom[d * H + hh] + 1e-16f);
        float4 hv = *(const float4*)(h + (size_t)s * F + j);     // b128 gather
        float* op = out + (size_t)d * F + j;
        atomicAdd(op + 0, hv.x * alpha);
        atomicAdd(op + 1, hv.y * alpha);
        atomicAdd(op + 2, hv.z * alpha);
        atomicAdd(op + 3, hv.w * alpha);
    }
}

// ============ act = elu(out1 + b1)  (float4) ============
__global__ void elu_bias4(const float* __restrict__ out1, const float* __restrict__ b,
                          float* __restrict__ act) {
    int idx = blockIdx.x * blockDim.x + threadIdx.x;     // over N*32 float4 groups
    if (idx >= N_NODES * (F1 / 4)) return;
    int j = (idx & (F1 / 4 - 1)) << 2;
    float4 v  = *(const float4*)(out1 + ((size_t)idx << 2));
    float4 bb = *(const float4*)(b + j);
    float4 r;
    r.x = (v.x + bb.x > 0.f) ? v.x + bb.x : __expf(v.x + bb.x) - 1.f;
    r.y = (v.y + bb.y > 0.f) ? v.y + bb.y : __expf(v.y + bb.y) - 1.f;
    r.z = (v.z + bb.z > 0.f) ? v.z + bb.z : __expf(v.z + bb.z) - 1.f;
    r.w = (v.w + bb.w > 0.f) ? v.w + bb.w : __expf(v.w + bb.w) - 1.f;
    *(float4*)(act + ((size_t)idx << 2)) = r;
}

// ============ final log_softmax(out2 + b2) ============
__global__ void log_softmax_k(const float* __restrict__ out2, const float* __restrict__ b,
                              float* __restrict__ y) {
    int n = blockIdx.x * blockDim.x + threadIdx.x;
    if (n >= N_NODES) return;
    float z[C2_];
    float mx = -INFINITY;
    #pragma unroll
    for (int c = 0; c < C2_; ++c) {
        z[c] = out2[(size_t)n * C2_ + c] + b[c];
        mx = fmaxf(mx, z[c]);
    }
    float s = 0.f;
    #pragma unroll
    for (int c = 0; c < C2_; ++c) s += __expf(z[c] - mx);
    float l = mx + __logf(s);
    #pragma unroll
    for (int c = 0; c < C2_; ++c) y[(size_t)n * C2_ + c] = z[c] - l;
}

static inline int nblk(long long work, int bs) { return (int)((work + bs - 1) / bs); }

extern "C" void kernel_launch(void* const* d_in, const int* in_sizes, int n_in,
                              void* d_out, int out_size, void* d_ws, size_t ws_size,
                              hipStream_t stream) {
    const float*     x    = (const float*)d_in[0];
    const long long* ei   = (const long long*)d_in[1];   // int64 [2,E]
    const float*     W1   = (const float*)d_in[2];
    const float*     asrc1= (const float*)d_in[3];
    const float*     adst1= (const float*)d_in[4];
    const float*     b1   = (const float*)d_in[5];
    const float*     W2   = (const float*)d_in[6];
    const float*     asrc2= (const float*)d_in[7];
    const float*     adst2= (const float*)d_in[8];
    const float*     b2   = (const float*)d_in[9];
    float*           y    = (float*)d_out;

    const int E  = in_sizes[1] / 2;
    const int ET = E + N_NODES;

    // ---- workspace carve-up (floats), ~74.5 MB total ----
    float* ws   = (float*)d_ws;
    float* h1   = ws;                                    // N*128
    float* as1  = h1  + (size_t)N_NODES * F1;            // N*8
    float* ad1  = as1 + (size_t)N_NODES * H1_;           // N*8
    unsigned* m1 = (unsigned*)(ad1 + (size_t)N_NODES * H1_);   // N*8
    float* den1 = (float*)(m1 + (size_t)N_NODES * H1_);  // N*8
    float* out1 = den1 + (size_t)N_NODES * H1_;          // N*128
    float* h2   = out1 + (size_t)N_NODES * F1;           // N*40
    float* as2  = h2  + (size_t)N_NODES * C2_;           // N
    float* ad2  = as2 + N_NODES;                         // N
    unsigned* m2 = (unsigned*)(ad2 + N_NODES);           // N
    float* den2 = (float*)(m2 + N_NODES);                // N
    float* out2 = den2 + N_NODES;                        // N*40

    // ---- zero / seed accumulators (capture-safe stream ops) ----
    hipMemsetAsync(den1, 0, (size_t)N_NODES * H1_ * sizeof(float), stream);
    hipMemsetAsync(out1, 0, (size_t)N_NODES * F1 * sizeof(float), stream);
    hipMemsetAsync(den2, 0, (size_t)N_NODES * sizeof(float), stream);
    hipMemsetAsync(out2, 0, (size_t)N_NODES * C2_ * sizeof(float), stream);
    init_u32<<<nblk(N_NODES * H1_, 256), 256, 0, stream>>>(m1, ORD_NEG_INF, N_NODES * H1_);
    init_u32<<<nblk(N_NODES, 256), 256, 0, stream>>>(m2, ORD_NEG_INF, N_NODES);

    // ---- layer 1 ----
    gemm1_wmma<<<nblk(N_NODES, 64), 256, 0, stream>>>(x, W1, h1);
    att_scores<<<nblk((long long)N_NODES * H1_, 256), 256, 0, stream>>>(
        h1, asrc1, adst1, as1, ad1, H1_, C1_);
    edge_max<<<nblk((long long)ET * H1_, 256), 256, 0, stream>>>(
        ei, E, ET, as1, ad1, m1, H1_);
    edge_expsum<<<nblk((long long)ET * H1_, 256), 256, 0, stream>>>(
        ei, E, ET, as1, ad1, m1, den1, H1_);
    edge_aggregate4<<<nblk((long long)ET * (F1 / 4), 256), 256, 0, stream>>>(
        ei, E, ET, as1, ad1, m1, den1, h1, out1, H1_, C1_);
    elu_bias4<<<nblk((long long)N_NODES * (F1 / 4), 256), 256, 0, stream>>>(out1, b1, h1);

    // ---- layer 2 ----
    gemm2_wmma<<<N_NODES / 16, 96, 0, stream>>>(h1, W2, h2);
    att_scores<<<nblk(N_NODES, 256), 256, 0, stream>>>(
        h2, asrc2, adst2, as2, ad2, 1, C2_);
    edge_max<<<nblk(ET, 256), 256, 0, stream>>>(ei, E, ET, as2, ad2, m2, 1);
    edge_expsum<<<nblk(ET, 256), 256, 0, stream>>>(ei, E, ET, as2, ad2, m2, den2, 1);
    edge_aggregate4<<<nblk((long long)ET * (C2_ / 4), 256), 256, 0, stream>>>(
        ei, E, ET, as2, ad2, m2, den2, h2, out2, 1, C2_);

    // ---- log_softmax ----
    log_softmax_k<<<nblk(N_NODES, 256), 256, 0, stream>>>(out2, b2, y);
}